// RWKV6Attention_62775241998771
// MI455X (gfx1250) — compile-verified
//
#include <hip/hip_runtime.h>
#include <hip/hip_bf16.h>

// ---------------- problem constants ----------------
#define BB   4
#define TT   2048
#define HID  1024
#define NH   4
#define KD   512
#define VD   1024
#define DK   128      // KD/NH
#define DV   256      // VD/NH
#define RR   32       // proj_low_rank_dim
#define GLR  64       // gate_low_rank_dim
#define MROWS (BB*TT) // 8192
#define EPSV 1e-5f

typedef __attribute__((ext_vector_type(2))) float v2f;
typedef __attribute__((ext_vector_type(8))) float v8f;

// ---- CDNA5 async load: global -> LDS, tracked by ASYNCcnt ----------------
__device__ __forceinline__ void async_ld_f32(void* lds, const float* g) {
  unsigned l = (unsigned)(uintptr_t)lds;   // low 32 bits of generic shared ptr = LDS offset
  asm volatile("global_load_async_to_lds_b32 %0, %1, off"
               :: "v"(l), "v"(g) : "memory");
}
__device__ __forceinline__ void wait_async0() {
  asm volatile("s_wait_asynccnt 0x0" ::: "memory");
}

__device__ __forceinline__ float epi_apply(int EPI, float v) {
  if (EPI == 1)      return tanhf(v);
  else if (EPI == 2) return __expf(-__expf(v));       // exp(w), w = -exp(raw)
  else if (EPI == 3) return v / (1.0f + __expf(-v));  // SiLU
  return v;
}

// ==========================================================================
// f32 WMMA GEMM, 16x16 tile per wave (used only for N=160 x_proj GEMM).
// ==========================================================================
template<int EPI>
__global__ __launch_bounds__(256)
void wmma_gemm_t16_kernel(const float* __restrict__ A,
                          const float* __restrict__ W,
                          const float* __restrict__ bias,
                          float* __restrict__ C,
                          int N, int K) {
  const int wave  = blockIdx.x * (blockDim.x >> 5) + (threadIdx.x >> 5);
  const int lane  = threadIdx.x & 31;
  const int ntile = N >> 4;
  const int total = (MROWS >> 4) * ntile;
  if (wave >= total) return;                 // wave-uniform

  const int mt = wave / ntile;
  const int nt = wave - mt * ntile;
  const int m0 = mt << 4, n0 = nt << 4;
  const int l15  = lane & 15;
  const int koff = (lane >> 4) << 1;

  const float* __restrict__ arow = A + (size_t)(m0 + l15) * K + koff;
  const float* __restrict__ wrow = W + (size_t)(n0 + l15) * K + koff;

  v8f acc = {};
  for (int k0 = 0; k0 < K; k0 += 4) {
    v2f a = *(const v2f*)(arow + k0);
    v2f b = *(const v2f*)(wrow + k0);
    acc = __builtin_amdgcn_wmma_f32_16x16x4_f32(false, a, false, b,
                                                (short)0, acc, false, false);
  }
  const int colg  = n0 + l15;
  const int mbase = m0 + ((lane >> 4) << 3);
  const float bv  = bias ? bias[colg] : 0.0f;
#pragma unroll
  for (int jj = 0; jj < 8; ++jj)
    C[(size_t)(mbase + jj) * N + colg] = epi_apply(EPI, acc[jj] + bv);
}

// ==========================================================================
// f32 WMMA GEMM, 16(M) x 64(N) per wave: A operand reused across 4 WMMAs.
// Requires N % 64 == 0.
// ==========================================================================
template<int EPI>
__global__ __launch_bounds__(256)
void wmma_gemm_t64_kernel(const float* __restrict__ A,
                          const float* __restrict__ W,
                          const float* __restrict__ bias,
                          float* __restrict__ C,
                          int N, int K) {
  const int wave  = blockIdx.x * (blockDim.x >> 5) + (threadIdx.x >> 5);
  const int lane  = threadIdx.x & 31;
  const int nt64  = N >> 6;
  const int total = (MROWS >> 4) * nt64;
  if (wave >= total) return;                 // wave-uniform

  const int mt = wave / nt64;
  const int nt = wave - mt * nt64;
  const int m0 = mt << 4, n0 = nt << 6;
  const int l15  = lane & 15;
  const int koff = (lane >> 4) << 1;

  const float* __restrict__ arow = A + (size_t)(m0 + l15) * K + koff;
  const float* __restrict__ w0   = W + (size_t)(n0 + l15) * K + koff;
  const size_t ws16 = (size_t)16 * K;        // W row stride for next 16-col subtile

  v8f acc[4] = {{}, {}, {}, {}};
  for (int k0 = 0; k0 < K; k0 += 4) {
    v2f a = *(const v2f*)(arow + k0);
#pragma unroll
    for (int s = 0; s < 4; ++s) {
      v2f b = *(const v2f*)(w0 + (size_t)s * ws16 + k0);
      acc[s] = __builtin_amdgcn_wmma_f32_16x16x4_f32(false, a, false, b,
                                                     (short)0, acc[s], false, false);
    }
  }
  const int mbase = m0 + ((lane >> 4) << 3);
#pragma unroll
  for (int s = 0; s < 4; ++s) {
    const int colg = n0 + (s << 4) + l15;
    const float bv = bias ? bias[colg] : 0.0f;
#pragma unroll
    for (int jj = 0; jj < 8; ++jj)
      C[(size_t)(mbase + jj) * N + colg] = epi_apply(EPI, acc[s][jj] + bv);
  }
}

// ==========================================================================
// lerp0: Abuf = x + (time_shift(x) - x) * mu0
// ==========================================================================
__global__ __launch_bounds__(256)
void lerp0_kernel(const float* __restrict__ x,
                  const float* __restrict__ mu0,
                  float* __restrict__ Abuf) {
  size_t idx = (size_t)blockIdx.x * blockDim.x + threadIdx.x;
  if (idx >= (size_t)MROWS * HID) return;
  int    i = (int)(idx & (HID - 1));
  size_t m = idx >> 10;
  int    t = (int)(m & (TT - 1));
  float xv = x[idx];
  float sh = (t == 0) ? 0.0f : x[idx - HID];
  Abuf[idx] = xv + (sh - xv) * mu0[i];
}

// ==========================================================================
// lerp_slot: Abuf[m,i] = x + (shift(x)-x) * (xp[m,slot,:] . W_x2[i,slot,:] + bias)
// ==========================================================================
__global__ __launch_bounds__(256)
void lerp_slot_kernel(const float* __restrict__ x,
                      const float* __restrict__ xp,
                      const float* __restrict__ W_x2,
                      const float* __restrict__ x_bias,
                      float* __restrict__ Abuf,
                      int slot) {
  __shared__ float sxp[RR];
  const int m = blockIdx.x;
  const int t = m & (TT - 1);
  if (threadIdx.x < RR)
    sxp[threadIdx.x] = xp[(size_t)m * (5 * RR) + slot * RR + threadIdx.x];
  __syncthreads();
  for (int i = threadIdx.x; i < HID; i += blockDim.x) {
    float mu = x_bias[slot * HID + i];
    const float* __restrict__ wrow = W_x2 + (size_t)i * (5 * RR) + slot * RR;
#pragma unroll
    for (int r = 0; r < RR; ++r) mu = fmaf(sxp[r], wrow[r], mu);
    size_t idx = (size_t)m * HID + i;
    float xv = x[idx];
    float sh = (t == 0) ? 0.0f : x[idx - HID];
    Abuf[idx] = xv + (sh - xv) * mu;
  }
}

// ==========================================================================
// RWKV6 recurrence, double-buffered via async global->LDS loads.
// Grid = B*NH*(DV/64) = 64 blocks x 256 threads. Thread: column j, k-quarter
// kq; S[k] for 32 k's lives in VGPRs. While step t computes, step t+1's
// r/k/decay/v are DMA'd into the alternate LDS buffers (ASYNCcnt pipeline).
// ==========================================================================
__global__ __launch_bounds__(256)
void rwkv6_scan_kernel(const float* __restrict__ rbuf,
                       const float* __restrict__ kbuf,
                       const float* __restrict__ dbuf,   // exp(w) decay
                       const float* __restrict__ vbuf,
                       const float* __restrict__ bonus,  // u: (NH, DK)
                       float* __restrict__ obuf) {
  const int blk = blockIdx.x;
  const int c   = blk & 3;          // column chunk (DV/64 = 4)
  const int h   = (blk >> 2) & 3;   // head
  const int b   = blk >> 4;         // batch
  const int tid = threadIdx.x;
  const int j   = c * 64 + (tid & 63);
  const int kq  = tid >> 6;         // 0..3

  __shared__ float rs[2][DK], ks[2][DK], dsh[2][DK], vsh[2][64];
  __shared__ float us[DK];
  __shared__ float red[256];

  if (tid < DK) us[tid] = bonus[h * DK + tid];

  float S[32];
#pragma unroll
  for (int kk = 0; kk < 32; ++kk) S[kk] = 0.0f;

  const int kcol0 = h * DK;
  const int vcol  = h * DV + j;
  const int kb    = kq * 32;

  // issue async stage of step t into LDS buffer pb
  auto issue = [&](int t, int pb) {
    const size_t rowK = ((size_t)(b * TT + t)) * KD + kcol0;
    const size_t rowV = ((size_t)(b * TT + t)) * VD + h * DV + c * 64;
    if (tid < DK) {
      async_ld_f32(&rs[pb][tid],  rbuf + rowK + tid);
      async_ld_f32(&dsh[pb][tid], dbuf + rowK + tid);
    } else {
      const int q = tid - DK;
      async_ld_f32(&ks[pb][q], kbuf + rowK + q);
      if (q < 64) async_ld_f32(&vsh[pb][q], vbuf + rowV + q);
    }
  };

  issue(0, 0);
  wait_async0();
  __syncthreads();                  // buf0 ready (also publishes us)

  for (int t = 0; t < TT; ++t) {
    const int cur = t & 1, nxt = cur ^ 1;
    if (t + 1 < TT) issue(t + 1, nxt);      // overlaps with compute below

    const float vj = vsh[cur][tid & 63];
    float acc = 0.0f;
#pragma unroll
    for (int kk = 0; kk < 32; ++kk) {
      const int ki = kb + kk;
      const float kv = ks[cur][ki] * vj;
      acc   = fmaf(rs[cur][ki], fmaf(us[ki], kv, S[kk]), acc);
      S[kk] = fmaf(dsh[cur][ki], S[kk], kv);
    }
    red[tid] = acc;
    __syncthreads();
    if (tid < 64) {
      float o = red[tid] + red[tid + 64] + red[tid + 128] + red[tid + 192];
      obuf[((size_t)(b * TT + t)) * VD + vcol] = o;   // (B,T,VD) layout
    }
    wait_async0();                  // next buffer filled by DMA
    __syncthreads();                // and everyone done with red/cur buffers
  }
}

// ==========================================================================
// GroupNorm (NH groups of DV) + affine + multiply by precomputed SiLU(g).
// ==========================================================================
__global__ __launch_bounds__(256)
void gn_gate_kernel(float* __restrict__ obuf,
                    const float* __restrict__ gbuf,
                    const float* __restrict__ gnw,
                    const float* __restrict__ gnb) {
  const int blk = blockIdx.x;
  const int h   = blk & 3;
  const int m   = blk >> 2;
  const size_t base = (size_t)m * VD + h * DV;
  const int tid = threadIdx.x;

  const float v = obuf[base + tid];
  __shared__ float s1[256], s2[256];
  s1[tid] = v; s2[tid] = v * v;
  __syncthreads();
  for (int off = 128; off > 0; off >>= 1) {
    if (tid < off) { s1[tid] += s1[tid + off]; s2[tid] += s2[tid + off]; }
    __syncthreads();
  }
  const float mean = s1[0] * (1.0f / DV);
  const float var  = s2[0] * (1.0f / DV) - mean * mean;
  const float nrm  = (v - mean) * rsqrtf(var + EPSV);
  const int col = h * DV + tid;
  obuf[base + tid] = fmaf(nrm, gnw[col], gnb[col]) * gbuf[base + tid];
}

// ==========================================================================
// Host side
// ==========================================================================
static inline dim3 grid_t16(int N) {
  int waves = (MROWS >> 4) * (N >> 4);
  return dim3((waves + 7) / 8);
}
static inline dim3 grid_t64(int N) {
  int waves = (MROWS >> 4) * (N >> 6);
  return dim3((waves + 7) / 8);
}

extern "C" void kernel_launch(void* const* d_in, const int* in_sizes, int n_in,
                              void* d_out, int out_size, void* d_ws, size_t ws_size,
                              hipStream_t stream) {
  (void)in_sizes; (void)n_in; (void)out_size; (void)ws_size;
  const float* x      = (const float*)d_in[0];
  const float* mu0    = (const float*)d_in[1];
  const float* W_x0   = (const float*)d_in[2];   // (160, 1024)
  const float* W_x2   = (const float*)d_in[3];   // (1024, 160)
  const float* x_bias = (const float*)d_in[4];   // (5, 1024)
  const float* W_r    = (const float*)d_in[5];   // (512, 1024)
  const float* W_w1   = (const float*)d_in[6];   // (64, 1024)
  const float* W_w2   = (const float*)d_in[7];   // (512, 64)
  const float* b_w2   = (const float*)d_in[8];   // (512)
  const float* W_k    = (const float*)d_in[9];   // (512, 1024)
  const float* W_v    = (const float*)d_in[10];  // (1024, 1024)
  const float* W_g    = (const float*)d_in[11];  // (1024, 1024)
  const float* bonus  = (const float*)d_in[12];  // (4, 128)
  const float* gnw    = (const float*)d_in[13];
  const float* gnb    = (const float*)d_in[14];
  const float* W_o    = (const float*)d_in[15];  // (1024, 1024)
  float* out = (float*)d_out;

  // -------- workspace carve-up (floats) --------
  float* ws   = (float*)d_ws;
  float* Abuf = ws;                               size_t off = (size_t)MROWS * HID;
  float* xp   = ws + off;                         off += (size_t)MROWS * 5 * RR;
  float* rbuf = ws + off;                         off += (size_t)MROWS * KD;
  float* dbuf = ws + off;                         off += (size_t)MROWS * KD;
  float* kbuf = ws + off;                         off += (size_t)MROWS * KD;
  float* vbuf = ws + off;                         off += (size_t)MROWS * VD;
  float* gbuf = ws + off;                         off += (size_t)MROWS * VD;
  float* w1b  = ws + off;                         off += (size_t)MROWS * GLR;
  float* obuf = Abuf;  // reuse: scan runs after the last GEMM that reads Abuf

  const dim3 blk256(256);

  // 1) x_proj input mix + GEMM + tanh -> xp (8192 x 160)
  {
    size_t n = (size_t)MROWS * HID;
    lerp0_kernel<<<dim3((unsigned)((n + 255) / 256)), blk256, 0, stream>>>(x, mu0, Abuf);
    wmma_gemm_t16_kernel<1><<<grid_t16(5 * RR), blk256, 0, stream>>>(Abuf, W_x0, nullptr, xp, 5 * RR, HID);
  }
  // 2) r  (slot 0)
  lerp_slot_kernel<<<dim3(MROWS), blk256, 0, stream>>>(x, xp, W_x2, x_bias, Abuf, 0);
  wmma_gemm_t64_kernel<0><<<grid_t64(KD), blk256, 0, stream>>>(Abuf, W_r, nullptr, rbuf, KD, HID);
  // 3) w  (slot 1): tanh LoRA down, up + bias, epilogue exp(-exp(.)) -> decay
  lerp_slot_kernel<<<dim3(MROWS), blk256, 0, stream>>>(x, xp, W_x2, x_bias, Abuf, 1);
  wmma_gemm_t64_kernel<1><<<grid_t64(GLR), blk256, 0, stream>>>(Abuf, W_w1, nullptr, w1b, GLR, HID);
  wmma_gemm_t64_kernel<2><<<grid_t64(KD), blk256, 0, stream>>>(w1b, W_w2, b_w2, dbuf, KD, GLR);
  // 4) k  (slot 2)
  lerp_slot_kernel<<<dim3(MROWS), blk256, 0, stream>>>(x, xp, W_x2, x_bias, Abuf, 2);
  wmma_gemm_t64_kernel<0><<<grid_t64(KD), blk256, 0, stream>>>(Abuf, W_k, nullptr, kbuf, KD, HID);
  // 5) v  (slot 3)
  lerp_slot_kernel<<<dim3(MROWS), blk256, 0, stream>>>(x, xp, W_x2, x_bias, Abuf, 3);
  wmma_gemm_t64_kernel<0><<<grid_t64(VD), blk256, 0, stream>>>(Abuf, W_v, nullptr, vbuf, VD, HID);
  // 6) g  (slot 4), epilogue SiLU
  lerp_slot_kernel<<<dim3(MROWS), blk256, 0, stream>>>(x, xp, W_x2, x_bias, Abuf, 4);
  wmma_gemm_t64_kernel<3><<<grid_t64(VD), blk256, 0, stream>>>(Abuf, W_g, nullptr, gbuf, VD, HID);
  // 7) recurrence -> obuf (= Abuf, now free); async-LDS double buffered
  rwkv6_scan_kernel<<<dim3(BB * NH * (DV / 64)), blk256, 0, stream>>>(rbuf, kbuf, dbuf, vbuf, bonus, obuf);
  // 8) GroupNorm + gate (in place)
  gn_gate_kernel<<<dim3(MROWS * NH), blk256, 0, stream>>>(obuf, gbuf, gnw, gnb);
  // 9) output projection -> d_out
  wmma_gemm_t64_kernel<0><<<grid_t64(HID), blk256, 0, stream>>>(obuf, W_o, nullptr, out, HID, VD);
}